// PositionEmbedding_51196010168778
// MI455X (gfx1250) — compile-verified
//
#include <hip/hip_runtime.h>
#include <hip/hip_bf16.h>

// out[b, r, d] = (r == 0 ? x_class[d] : inputs[b, r-1, d]) + emb[r, d]
// B=64, N=1024, D=768 (fp32).  Pure streaming op: ~404 MB HBM traffic,
// ~17.4 us roofline at 23.3 TB/s on MI455X.  No contraction dim -> matrix
// pipes (WMMA) are inapplicable; the optimized kernel is defined by its
// data-movement mix: b128 vector memops, NT cache hints on the streamed
// arrays (inputs read once, out written once), temporal hint on the 3 MB
// embedding table so it stays L2-resident, and a 4-wide batch tile so each
// emb element is loaded once and reused from registers.

typedef __attribute__((ext_vector_type(4))) float v4f;

#define BATCH_TILE 4

template <bool GUARDED>
__global__ __launch_bounds__(192)
void pos_embed_add_kernel(const v4f* __restrict__ inputs,   // [B, N, D/4]
                          const v4f* __restrict__ xcls,     // [D/4]
                          const v4f* __restrict__ emb,      // [N+1, D/4]
                          v4f* __restrict__ out,            // [B, N+1, D/4]
                          int B, int N, int D4) {
  const int d4 = threadIdx.x;          // 0 .. D4-1  (D4 = 192 for D=768)
  const int r  = blockIdx.x;           // 0 .. N     (output row)
  const int b0 = blockIdx.y * BATCH_TILE;

  // Embedding element for (r, d4): shared by all batches in this tile.
  // Temporal load -> hot in L2 across the batch-tile blocks with same r.
  const v4f e = emb[(size_t)r * D4 + d4];

  if (r == 0) {
    // Class token row: same value for every batch image.
    const v4f c = xcls[d4];
    const v4f v = c + e;
#pragma unroll
    for (int i = 0; i < BATCH_TILE; ++i) {
      const int b = b0 + i;
      if (!GUARDED || b < B) {
        __builtin_nontemporal_store(
            v, &out[((size_t)b * (N + 1)) * D4 + d4]);
      }
    }
  } else {
    // Issue all tile loads first (4 outstanding b128 NT loads / thread to
    // cover HBM latency), then add, then NT stores.  Straight-line code in
    // the unguarded instantiation.
    v4f x[BATCH_TILE];
#pragma unroll
    for (int i = 0; i < BATCH_TILE; ++i) {
      const int b = b0 + i;
      if (!GUARDED || b < B) {
        x[i] = __builtin_nontemporal_load(
            &inputs[((size_t)b * N + (r - 1)) * D4 + d4]);
      }
    }
#pragma unroll
    for (int i = 0; i < BATCH_TILE; ++i) {
      const int b = b0 + i;
      if (!GUARDED || b < B) {
        const v4f v = x[i] + e;
        __builtin_nontemporal_store(
            v, &out[((size_t)b * (N + 1) + r) * D4 + d4]);
      }
    }
  }
}

extern "C" void kernel_launch(void* const* d_in, const int* in_sizes, int n_in,
                              void* d_out, int out_size, void* d_ws, size_t ws_size,
                              hipStream_t stream) {
  // setup_inputs() order: inputs [B,N,D], x_class [D], embeddings [N+1,D]
  const v4f* inputs = (const v4f*)d_in[0];
  const v4f* xcls   = (const v4f*)d_in[1];
  const v4f* emb    = (const v4f*)d_in[2];
  v4f*       out    = (v4f*)d_out;

  const int D  = in_sizes[1];               // 768
  const int N  = in_sizes[2] / D - 1;       // 1024
  const int B  = in_sizes[0] / (N * D);     // 64
  const int D4 = D / 4;                     // 192 threads = 6 waves (wave32)

  dim3 grid(N + 1, (B + BATCH_TILE - 1) / BATCH_TILE);  // (1025, 16)
  dim3 block(D4);

  if (B % BATCH_TILE == 0) {
    // Exact tiling (the B=64 case): branch-free hot path.
    pos_embed_add_kernel<false><<<grid, block, 0, stream>>>(
        inputs, xcls, emb, out, B, N, D4);
  } else {
    pos_embed_add_kernel<true><<<grid, block, 0, stream>>>(
        inputs, xcls, emb, out, B, N, D4);
  }
}